// Attention_2095944040732
// MI455X (gfx1250) — compile-verified
//
#include <hip/hip_runtime.h>

typedef __attribute__((ext_vector_type(2))) float v2f;
typedef __attribute__((ext_vector_type(4))) float v4f;
typedef __attribute__((ext_vector_type(8))) float v8f;

#define WMMA_F32(a, b, c) \
  __builtin_amdgcn_wmma_f32_16x16x4_f32(false, (a), false, (b), (short)0, (c), false, false)

constexpr int BATCH = 16, CIN = 512, SEQ = 1024, MOUT = 1536, DH = 128, NHEADS = 4;
constexpr float SCALE = 0.08838834764831845f;  // 128^-0.5

// ---- gfx1250 async global->LDS copy (ASYNCcnt-tracked DMA, no VGPR bounce) ----
// Inline asm per CDNA5 ISA 15.18.3 (GLOBAL_LOAD_ASYNC_TO_LDS_B128): vdst is the
// per-lane LDS byte offset, vaddr the 64-bit global address, no SADDR ("off").
__device__ __forceinline__ void async_copy16(const float* g, float* l) {
  unsigned lds_off =
      (unsigned)(unsigned long long)(__attribute__((address_space(3))) void*)l;
  asm volatile("global_load_async_to_lds_b128 %0, %1, off"
               :: "v"(lds_off), "v"(g) : "memory");
}

#if __has_builtin(__builtin_amdgcn_s_wait_asynccnt)
#define WAIT_ASYNC(n) __builtin_amdgcn_s_wait_asynccnt(n)
#else
#define WAIT_ASYNC(n) asm volatile("s_wait_asynccnt %0" :: "i"(n) : "memory")
#endif

#define WAIT_DS0() asm volatile("s_wait_dscnt 0x0" ::: "memory")

// ---------------------------------------------------------------------------
// Kernel A: Y[b] = w_qkv (1536x512) x fmap[b] (512x1024), double-buffered
// async-LDS staging. Workgroup tile 128x128, 8 waves as 4x2, wave tile 32x64
// (2x4 WMMA accumulators -> 8 wmma per 6 fragment loads per k-step).
// Epilogue: Q rows pre-scaled by 128^-0.5; K rows get emb folded in.
// ---------------------------------------------------------------------------
__global__ __launch_bounds__(256) void qkv_gemm_kernel(
    const float* __restrict__ fmap, const float* __restrict__ w,
    const float* __restrict__ height, const float* __restrict__ width,
    float* __restrict__ Y) {
  __shared__ float Ws[2][128][36];   // 128x32 W tile  (row 144B, 16B-aligned)
  __shared__ float Xs[2][32][132];   // 32x128 X tile  (row 528B, 16B-aligned)

  const int tid = threadIdx.x;
  const int lane = tid & 31, wid = tid >> 5;
  const int hl = lane >> 4, ln = lane & 15;
  const int wm = wid >> 1, wn = wid & 1;  // 4(M) x 2(N) wave grid
  const int b = blockIdx.z;
  const int m0 = blockIdx.y * 128;
  const int n0 = blockIdx.x * 128;
  const float* X = fmap + (size_t)b * CIN * SEQ;

  auto stage = [&](int kb, int bb) {  // 8 async b128 per thread per tile
#pragma unroll
    for (int it = 0; it < 4; ++it) {
      int fi = tid + it * 256;                      // 1024 x 16B  (W 128x32)
      int r = fi >> 3, c4 = (fi & 7) << 2;
      async_copy16(&w[(size_t)(m0 + r) * CIN + kb + c4], &Ws[bb][r][c4]);
    }
#pragma unroll
    for (int it = 0; it < 4; ++it) {
      int fi = tid + it * 256;                      // 1024 x 16B  (X 32x128)
      int r = fi >> 5, c4 = (fi & 31) << 2;
      async_copy16(&X[(size_t)(kb + r) * SEQ + n0 + c4], &Xs[bb][r][c4]);
    }
  };

  v8f acc[2][4] = {};
  stage(0, 0);
  for (int t = 0; t < CIN / 32; ++t) {
    const int cur = t & 1;
    if (t < CIN / 32 - 1) {
      stage((t + 1) * 32, cur ^ 1);   // prefetch next tile
      WAIT_ASYNC(8);                  // in-order completion -> tile t landed
    } else {
      WAIT_ASYNC(0);
    }
    __syncthreads();

#pragma unroll
    for (int kk = 0; kk < 32; kk += 4) {
      int krow = kk + 2 * hl;  // this lane holds K = krow, krow+1
      v2f af[2], bf[4];
#pragma unroll
      for (int mt = 0; mt < 2; ++mt) {
        int row = wm * 32 + mt * 16 + ln;
        af[mt].x = Ws[cur][row][krow];
        af[mt].y = Ws[cur][row][krow + 1];
      }
#pragma unroll
      for (int nt = 0; nt < 4; ++nt) {
        int col = wn * 64 + nt * 16 + ln;
        bf[nt].x = Xs[cur][krow][col];
        bf[nt].y = Xs[cur][krow + 1][col];
      }
#pragma unroll
      for (int mt = 0; mt < 2; ++mt)
#pragma unroll
        for (int nt = 0; nt < 4; ++nt)
          acc[mt][nt] = WMMA_F32(af[mt], bf[nt], acc[mt][nt]);
    }
    __syncthreads();
  }

#pragma unroll
  for (int mt = 0; mt < 2; ++mt)
#pragma unroll
    for (int nt = 0; nt < 4; ++nt)
#pragma unroll
      for (int r = 0; r < 8; ++r) {
        int o = m0 + wm * 32 + mt * 16 + r + 8 * hl;
        int j = n0 + wn * 64 + nt * 16 + ln;
        float v = acc[mt][nt][r];
        if (o < 512) {
          v *= SCALE;
        } else if (o < 1024) {
          int d = o & 127;
          v += height[(j >> 5) * DH + d] + width[(j & 31) * DH + d];
        }
        Y[((size_t)b * MOUT + o) * SEQ + j] = v;
      }
}

// ---------------------------------------------------------------------------
// Kernel B: flash attention per (b,h); 128 query rows per workgroup (8 waves
// x 16 rows). Q frags live in VGPRs; K double-buffered via async-LDS DMA; V
// DMA overlapped with the S-GEMM via a split asynccnt wait; online softmax
// state in registers; P re-laid out C->A via per-wave LDS; output transposed
// through LDS for float4 stores.
// ---------------------------------------------------------------------------
__global__ __launch_bounds__(256) void flash_attn_kernel(
    const float* __restrict__ Y, float* __restrict__ out) {
  __shared__ float Ks[2][128][68];
  __shared__ float Vs[128][68];
  __shared__ float Ps[8][1088];  // per-wave: P (16x64 @ stride 68) / O chunk (32x20)

  const int tid = threadIdx.x;
  const int lane = tid & 31, wid = tid >> 5;
  const int hl = lane >> 4, ln = lane & 15;
  const int h = blockIdx.y, b = blockIdx.z;
  const int i0 = blockIdx.x * 128 + wid * 16;

  const float* Q  = Y + ((size_t)b * MOUT + h * DH) * SEQ;
  const float* Ke = Y + ((size_t)b * MOUT + 512 + h * DH) * SEQ;
  const float* V  = Y + ((size_t)b * MOUT + 1024 + h * DH) * SEQ;

  auto stageK = [&](int j0, int bb) {  // 8 async b128 per thread
#pragma unroll
    for (int it = 0; it < 8; ++it) {
      int fi = tid + it * 256;
      int r = fi >> 4, c4 = (fi & 15) << 2;
      async_copy16(&Ke[(size_t)r * SEQ + j0 + c4], &Ks[bb][r][c4]);
    }
  };
  auto stageV = [&](int j0) {
#pragma unroll
    for (int it = 0; it < 8; ++it) {
      int fi = tid + it * 256;
      int r = fi >> 4, c4 = (fi & 15) << 2;
      async_copy16(&V[(size_t)r * SEQ + j0 + c4], &Vs[r][c4]);
    }
  };

  // Q A-fragments: 16(i) x 128(d); K-dim = d, 32 k-steps of 4.
  v2f qa[32];
#pragma unroll
  for (int ks = 0; ks < 32; ++ks) {
    int d = ks * 4 + 2 * hl;
    qa[ks].x = Q[(size_t)d * SEQ + i0 + ln];
    qa[ks].y = Q[(size_t)(d + 1) * SEQ + i0 + ln];
  }

  float rowm[8], rowl[8];
  v8f oacc[8] = {};
#pragma unroll
  for (int r = 0; r < 8; ++r) { rowm[r] = -1e30f; rowl[r] = 0.0f; }

  float* Pw = &Ps[wid][0];

  stageK(0, 0);
  for (int jt = 0; jt < SEQ / 64; ++jt) {
    const int j0 = jt * 64;
    const int cur = jt & 1;
    const bool pf = (jt < SEQ / 64 - 1);

    stageV(j0);                          // V buffer free (barrier at loop end)
    if (pf) { stageK(j0 + 64, cur ^ 1); WAIT_ASYNC(16); }  // K(jt) landed
    else    { WAIT_ASYNC(8); }
    __syncthreads();                     // K(jt) visible workgroup-wide

    // S (16x64) = Q_i^T x Ke_j  (Q carries the softmax scale)
    v8f s[4] = {};
#pragma unroll
    for (int ks = 0; ks < 32; ++ks) {
      int krow = ks * 4 + 2 * hl;
#pragma unroll
      for (int nt = 0; nt < 4; ++nt) {
        v2f bf;
        bf.x = Ks[cur][krow][nt * 16 + ln];
        bf.y = Ks[cur][krow + 1][nt * 16 + ln];
        s[nt] = WMMA_F32(qa[ks], bf, s[nt]);
      }
    }

    // Online softmax (rows owned by this half-wave: m = r + 8*hl).
#pragma unroll
    for (int r = 0; r < 8; ++r) {
      float mx = fmaxf(fmaxf(s[0][r], s[1][r]), fmaxf(s[2][r], s[3][r]));
#pragma unroll
      for (int off = 1; off < 16; off <<= 1) mx = fmaxf(mx, __shfl_xor(mx, off));
      float mnew = fmaxf(rowm[r], mx);
      float alpha = __expf(rowm[r] - mnew);
      rowm[r] = mnew;
      float psum = 0.0f;
#pragma unroll
      for (int nt = 0; nt < 4; ++nt) {
        float p = __expf(s[nt][r] - mnew);
        s[nt][r] = p;
        psum += p;
      }
#pragma unroll
      for (int off = 1; off < 16; off <<= 1) psum += __shfl_xor(psum, off);
      rowl[r] = rowl[r] * alpha + psum;
#pragma unroll
      for (int dt = 0; dt < 8; ++dt) oacc[dt][r] *= alpha;
    }

    // C-layout P -> A-layout via per-wave LDS scratch.
#pragma unroll
    for (int nt = 0; nt < 4; ++nt)
#pragma unroll
      for (int r = 0; r < 8; ++r)
        Pw[(r + 8 * hl) * 68 + nt * 16 + ln] = s[nt][r];
    WAIT_DS0();

    if (pf) WAIT_ASYNC(8); else WAIT_ASYNC(0);  // V(jt) landed (K(jt+1) may fly)
    __syncthreads();                            // V(jt) visible workgroup-wide

    // O (16x128) += P (16x64) x V^T; B[k=j][n=d] = Vs[d][j].
#pragma unroll
    for (int ks = 0; ks < 16; ++ks) {
      int kb = ks * 4 + 2 * hl;
      v2f af;
      af.x = Pw[ln * 68 + kb];
      af.y = Pw[ln * 68 + kb + 1];
#pragma unroll
      for (int dt = 0; dt < 8; ++dt) {
        v2f bf;
        bf.x = Vs[dt * 16 + ln][kb];
        bf.y = Vs[dt * 16 + ln][kb + 1];
        oacc[dt] = WMMA_F32(af, bf, oacc[dt]);
      }
    }
    __syncthreads();  // Vs/Ps consumers done before next iteration's DMA
  }

  // Normalize rows.
#pragma unroll
  for (int r = 0; r < 8; ++r) {
    float inv = 1.0f / rowl[r];
#pragma unroll
    for (int dt = 0; dt < 8; ++dt) oacc[dt][r] *= inv;
  }

  // Transpose 16(i) x 128(d) through LDS in 32-channel chunks -> float4 stores.
  float* obase = out + (size_t)(b * NHEADS + h) * DH * SEQ;
#pragma unroll
  for (int q = 0; q < 4; ++q) {
    WAIT_DS0();
#pragma unroll
    for (int dt = 0; dt < 2; ++dt)
#pragma unroll
      for (int r = 0; r < 8; ++r)
        Pw[(dt * 16 + ln) * 20 + r + 8 * hl] = oacc[q * 2 + dt][r];
    WAIT_DS0();
    {
      int c = lane;             // 32 lanes -> 32 channels per chunk
      int ch = q * 32 + c;
#pragma unroll
      for (int q4 = 0; q4 < 4; ++q4) {
        v4f tv = *(const v4f*)&Pw[c * 20 + q4 * 4];
        *(v4f*)&obase[(size_t)ch * SEQ + i0 + q4 * 4] = tv;
      }
    }
  }
}

extern "C" void kernel_launch(void* const* d_in, const int* in_sizes, int n_in,
                              void* d_out, int out_size, void* d_ws, size_t ws_size,
                              hipStream_t stream) {
  const float* fmap   = (const float*)d_in[0];
  const float* w_qkv  = (const float*)d_in[1];
  const float* height = (const float*)d_in[2];
  const float* width  = (const float*)d_in[3];
  float* out = (float*)d_out;
  float* Y = (float*)d_ws;  // 16 * 1536 * 1024 f32 = ~100.7 MB scratch

  dim3 gridA(SEQ / 128, MOUT / 128, BATCH);
  qkv_gemm_kernel<<<gridA, 256, 0, stream>>>(fmap, w_qkv, height, width, Y);

  dim3 gridB(SEQ / 128, NHEADS, BATCH);
  flash_attn_kernel<<<gridB, 256, 0, stream>>>(Y, out);
}